// SimpleSaccadeClassifier_64965675320093
// MI455X (gfx1250) — compile-verified
//
#include <hip/hip_runtime.h>

#define B_ 128
#define T_ 2048
#define D_ 128
#define H_ 64
#define O_ 8
#define F_ (3 * D_) /* 384 */

typedef __attribute__((ext_vector_type(2))) float v2f;
typedef __attribute__((ext_vector_type(8))) float v8f;

// ---------------------------------------------------------------------------
// Kernel 1: masked mean / unbiased std / last-row gather -> feats [B, 384]
// One block per batch row. 512 threads = 32 float4-channel groups x 16 time
// slices. Fully coalesced b128 streaming loads of x (134 MB total, HBM-bound).
// ---------------------------------------------------------------------------
__global__ __launch_bounds__(512) void saccade_feats_kernel(
    const float* __restrict__ x, const int* __restrict__ lengths,
    float* __restrict__ feats) {
  __shared__ float4 s_sum[512];
  __shared__ float4 s_sq[512];

  const int b = blockIdx.x;
  const int tid = (int)threadIdx.x;
  const int dg = tid & 31;     // float4 group: channels 4*dg .. 4*dg+3
  const int slice = tid >> 5;  // time slice 0..15
  const int len = lengths[b];
  const float* __restrict__ xb = x + (size_t)b * (size_t)(T_ * D_);

  float s0 = 0.f, s1 = 0.f, s2 = 0.f, s3 = 0.f;
  float q0 = 0.f, q1 = 0.f, q2 = 0.f, q3 = 0.f;
  for (int t = slice; t < len; t += 16) {
    const float4 v = ((const float4*)(xb + (size_t)t * D_))[dg];
    s0 += v.x; q0 += v.x * v.x;
    s1 += v.y; q1 += v.y * v.y;
    s2 += v.z; q2 += v.z * v.z;
    s3 += v.w; q3 += v.w * v.w;
  }
  s_sum[tid] = make_float4(s0, s1, s2, s3);
  s_sq[tid]  = make_float4(q0, q1, q2, q3);
  __syncthreads();

  if (slice == 0) {
    float4 S = s_sum[dg];
    float4 Q = s_sq[dg];
#pragma unroll
    for (int i = 1; i < 16; ++i) {
      const float4 a = s_sum[dg + 32 * i];
      const float4 c = s_sq[dg + 32 * i];
      S.x += a.x; S.y += a.y; S.z += a.z; S.w += a.w;
      Q.x += c.x; Q.y += c.y; Q.z += c.z; Q.w += c.w;
    }
    const float cnt = (float)len;
    const float invc = 1.0f / cnt;
    const float invc1 = 1.0f / (cnt - 1.0f);

    float mean[4], stdv[4];
    const float Sa[4] = {S.x, S.y, S.z, S.w};
    const float Qa[4] = {Q.x, Q.y, Q.z, Q.w};
#pragma unroll
    for (int j = 0; j < 4; ++j) {
      const float m = Sa[j] * invc;
      mean[j] = m;
      float var = (Qa[j] - Sa[j] * m) * invc1;  // == (sum_sq - sum^2/n)/(n-1)
      var = var > 0.f ? var : 0.f;
      stdv[j] = sqrtf(var);
    }
    const float4 L = ((const float4*)(xb + (size_t)(len - 1) * D_))[dg];
    const float last[4] = {L.x, L.y, L.z, L.w};

    float* __restrict__ fb = feats + (size_t)b * F_;
#pragma unroll
    for (int j = 0; j < 4; ++j) {
      const int d = 4 * dg + j;
      fb[d] = mean[j];
      fb[D_ + d] = stdv[j];
      fb[2 * D_ + d] = last[j];
    }
  }
}

// ---------------------------------------------------------------------------
// Kernel 2: MLP with V_WMMA_F32_16X16X4_F32 (exact f32 matrix math).
// Single 1024-thread block (32 waves).
//   GEMM1: feats[128,384] @ W1[384,64] + b1, ReLU -> h in LDS (32 KB)
//          32 waves = 8 M-tiles x 4 N-tiles of 16x16, K = 384 in steps of 4.
//   GEMM2: h[128,64] @ W2[64,8] + b2 -> out[128,8]
//          8 waves, N padded 8 -> 16 with zero B columns.
// ---------------------------------------------------------------------------
__global__ __launch_bounds__(1024) void saccade_mlp_kernel(
    const float* __restrict__ feats, const float* __restrict__ W1,
    const float* __restrict__ b1, const float* __restrict__ W2,
    const float* __restrict__ b2, float* __restrict__ out) {
  __shared__ float hLDS[B_ * H_];  // 128 x 64 f32 = 32 KB

  const int tid = (int)threadIdx.x;
  const int lane = tid & 31;
  const int wave = tid >> 5;       // 0..31
  const int halfsel = lane >> 4;   // 0: K pair {0,1}/M rows 0-7; 1: {2,3}/8-15
  const int lane16 = lane & 15;

  // ---------------- GEMM1 ----------------
  {
    const int m0 = (wave >> 2) * 16;  // 8 M tiles
    const int n0 = (wave & 3) * 16;   // 4 N tiles
    const int col = n0 + lane16;      // this lane's output column

    v8f acc;
    const float bias = b1[col];
#pragma unroll
    for (int r = 0; r < 8; ++r) acc[r] = bias;

    const int rowA = m0 + lane16;
    const float* __restrict__ arow = feats + (size_t)rowA * F_ + (halfsel << 1);

    for (int k0 = 0; k0 < F_; k0 += 4) {
      v2f a, bm;
      // A 16x4 f32 fragment: lanes 0-15 -> K = k0+0,k0+1 ; lanes 16-31 -> k0+2,k0+3
      a.x = arow[k0];
      a.y = arow[k0 + 1];
      // B 4x16 f32 fragment: mirrored layout over N
      const int kb = k0 + (halfsel << 1);
      bm.x = W1[(size_t)kb * H_ + col];
      bm.y = W1[(size_t)(kb + 1) * H_ + col];
      acc = __builtin_amdgcn_wmma_f32_16x16x4_f32(
          /*neg_a=*/false, a, /*neg_b=*/false, bm,
          /*c_mod=*/(short)0, acc, /*reuse_a=*/false, /*reuse_b=*/false);
    }

#pragma unroll
    for (int r = 0; r < 8; ++r) {
      const float v = acc[r] > 0.f ? acc[r] : 0.f;  // ReLU
      const int m = m0 + r + (halfsel << 3);
      hLDS[m * H_ + col] = v;
    }
  }
  __syncthreads();

  // ---------------- GEMM2 ----------------
  if (wave < 8) {  // wave-uniform branch: EXEC all-1s inside (WMMA-legal)
    const int m0 = wave * 16;
    const int col = lane16;          // 0..15, valid output cols are < 8
    const bool colOK = col < O_;

    v8f acc;
    const float bias = colOK ? b2[col] : 0.f;
#pragma unroll
    for (int r = 0; r < 8; ++r) acc[r] = bias;

    const int rowA = m0 + lane16;
#pragma unroll
    for (int k0 = 0; k0 < H_; k0 += 4) {
      const int kb = k0 + (halfsel << 1);
      v2f a, bm;
      a.x = hLDS[rowA * H_ + kb];
      a.y = hLDS[rowA * H_ + kb + 1];
      bm.x = colOK ? W2[(size_t)kb * O_ + col] : 0.f;
      bm.y = colOK ? W2[(size_t)(kb + 1) * O_ + col] : 0.f;
      acc = __builtin_amdgcn_wmma_f32_16x16x4_f32(
          false, a, false, bm, (short)0, acc, false, false);
    }

    if (colOK) {
#pragma unroll
      for (int r = 0; r < 8; ++r) {
        const int m = m0 + r + (halfsel << 3);
        out[m * O_ + col] = acc[r];
      }
    }
  }
}

extern "C" void kernel_launch(void* const* d_in, const int* in_sizes, int n_in,
                              void* d_out, int out_size, void* d_ws,
                              size_t ws_size, hipStream_t stream) {
  (void)in_sizes; (void)n_in; (void)out_size; (void)ws_size;
  const float* x = (const float*)d_in[0];
  const int* lengths = (const int*)d_in[1];
  const float* W1 = (const float*)d_in[2];
  const float* b1 = (const float*)d_in[3];
  const float* W2 = (const float*)d_in[4];
  const float* b2 = (const float*)d_in[5];
  float* out = (float*)d_out;
  float* feats = (float*)d_ws;  // [128, 384] f32 = 192 KB scratch

  saccade_feats_kernel<<<dim3(B_), dim3(512), 0, stream>>>(x, lengths, feats);
  saccade_mlp_kernel<<<dim3(1), dim3(1024), 0, stream>>>(feats, W1, b1, W2, b2,
                                                         out);
}